// MyModel_61933428415739
// MI455X (gfx1250) — compile-verified
//
#include <hip/hip_runtime.h>
#include <math.h>

typedef __attribute__((ext_vector_type(2))) float v2f;
typedef __attribute__((ext_vector_type(8))) float v8f;

#define N_PTS     2048
#define KNB       5
#define TOPK_N    512
#define INV_S2    100.0f      // 1/sigma^2, sigma=0.1
#define CAND_CAP  65536
#define FLATK     (N_PTS * KNB * KNB)   // 51200

// ---------------------------------------------------------------------------
// KNN via V_WMMA_F32_16X16X4_F32.
// A row m  = [p0, p1, p2, 1]          (lane layout: lanes 0-15 K=0,1; 16-31 K=2,3)
// B col n  = [-2q0, -2q1, -2q2, |q|^2]
// => C[m][n] = -2 p.q + |q|^2 ;  d2 = C + |p|^2 (lane-private row norm).
// One wave per 16-row band, 128 column tiles, per-row top-5 kept in registers
// by lanes 0-15 (strict < scan in ascending column order == JAX tie order).
// ---------------------------------------------------------------------------
__global__ __launch_bounds__(32) void knn_wmma_kernel(const float* __restrict__ pts,
                                                      int* __restrict__ ne_idx,
                                                      float* __restrict__ ne_dist) {
  __shared__ float tile[16 * 16];
  const int lane = threadIdx.x;
  const int lo = lane & 15;
  const int hi = lane >> 4;
  const int row = blockIdx.x * 16 + lo;

  const float px = pts[row * 3 + 0];
  const float py = pts[row * 3 + 1];
  const float pz = pts[row * 3 + 2];
  const float pn2 = px * px + py * py + pz * pz;

  v2f a;
  if (hi == 0) { a.x = px; a.y = py; } else { a.x = pz; a.y = 1.0f; }

  float bd[KNB];
  int   bi[KNB];
#pragma unroll
  for (int k = 0; k < KNB; ++k) { bd[k] = 3.4e38f; bi[k] = 0; }

  for (int t = 0; t < N_PTS / 16; ++t) {
    const int col = t * 16 + lo;
    const float qx = pts[col * 3 + 0];
    const float qy = pts[col * 3 + 1];
    const float qz = pts[col * 3 + 2];
    const float qn2 = qx * qx + qy * qy + qz * qz;
    v2f b;
    if (hi == 0) { b.x = -2.0f * qx; b.y = -2.0f * qy; }
    else         { b.x = -2.0f * qz; b.y = qn2; }

    v8f c = {};
    // 8 args: (neg_a, A, neg_b, B, c_mod, C, reuse_a, reuse_b)
    c = __builtin_amdgcn_wmma_f32_16x16x4_f32(false, a, false, b, (short)0, c,
                                              false, false);
#pragma unroll
    for (int v = 0; v < 8; ++v)
      tile[(v + 8 * hi) * 16 + lo] = c[v];
    __syncthreads();

    if (lane < 16) {
#pragma unroll 4
      for (int n = 0; n < 16; ++n) {
        const int cidx = t * 16 + n;
        if (cidx == row) continue;           // exclude self (reference drops it)
        float d = tile[lo * 16 + n] + pn2;
        int ci = cidx;
        if (d < bd[KNB - 1]) {
#pragma unroll
          for (int k = 0; k < KNB; ++k) {    // sorted bubble-insert (registers)
            bool sw = d < bd[k];
            float td = bd[k]; int ti = bi[k];
            bd[k] = sw ? d : td;  bi[k] = sw ? ci : ti;
            d     = sw ? td : d;  ci    = sw ? ti : ci;
          }
        }
      }
    }
    __syncthreads();
  }

  if (lane < 16) {
#pragma unroll
    for (int k = 0; k < KNB; ++k) {
      const int idx = bi[k];
      ne_idx[row * KNB + k] = idx;
      const float dx = px - pts[idx * 3 + 0];
      const float dy = py - pts[idx * 3 + 1];
      const float dz = pz - pts[idx * 3 + 2];
      ne_dist[row * KNB + k] = sqrtf(dx * dx + dy * dy + dz * dz);
    }
  }
}

// ---------------------------------------------------------------------------
// Workspace init: zero histogram (65536) + counter + cut word.
// ---------------------------------------------------------------------------
__global__ void init_ws_kernel(unsigned* __restrict__ p, int n) {
  const int g = blockIdx.x * blockDim.x + threadIdx.x;
  if (g < n) p[g] = 0u;
}

// ---------------------------------------------------------------------------
// Pass 1: histogram of float-bit keys (values are all >= 0, so raw bits are
// monotone). Bucket = bits >> 16 (exp + 7 mantissa bits). Zeros skipped.
// ---------------------------------------------------------------------------
__global__ __launch_bounds__(256) void pair_hist_kernel(
    const float* __restrict__ aff,
    const int* __restrict__ sidx, const float* __restrict__ sdist,
    const int* __restrict__ ridx, const float* __restrict__ rdist,
    unsigned* __restrict__ hist) {
  const int gid = blockIdx.x * 256 + threadIdx.x;
  const int s = gid >> 11;
  const int r = gid & (N_PTS - 1);
  const float base = aff[gid];

  float sd[KNB]; int so[KNB]; float rd[KNB]; int rj[KNB];
#pragma unroll
  for (int i = 0; i < KNB; ++i) {
    sd[i] = sdist[s * KNB + i];
    so[i] = sidx[s * KNB + i] << 11;     // row offset into affinity
    rd[i] = rdist[r * KNB + i];
    rj[i] = ridx[r * KNB + i];
  }
  if (base <= 0.0f) return;
#pragma unroll
  for (int i = 0; i < KNB; ++i) {
#pragma unroll
    for (int j = 0; j < KNB; ++j) {
      const float del = sd[i] - rd[j];
      const float tmp = 1.0f - del * del * INV_S2;
      if (tmp > 0.0f) {
        const float v = base * tmp * aff[so[i] + rj[j]];
        if (v > 0.0f) atomicAdd(&hist[__float_as_uint(v) >> 16], 1u);
      }
    }
  }
}

// ---------------------------------------------------------------------------
// Suffix-scan histogram from the top to find the bucket where the cumulative
// count reaches TOPK; write its lower-edge bit pattern as the collect cut.
// ---------------------------------------------------------------------------
__global__ __launch_bounds__(1024) void hist_scan_kernel(
    const unsigned* __restrict__ hist, unsigned* __restrict__ cutp) {
  __shared__ unsigned csum[1024];
  const int t = threadIdx.x;
  unsigned ssum = 0;
#pragma unroll 8
  for (int k = 0; k < 64; ++k) ssum += hist[t * 64 + k];
  csum[t] = ssum;
  __syncthreads();
  if (t == 0) {
    unsigned acc = 0;
    int chunk = 1023;
    for (; chunk >= 0; --chunk) {
      if (acc + csum[chunk] >= (unsigned)TOPK_N) break;
      acc += csum[chunk];
    }
    unsigned cut = 1u;                    // fallback: collect every positive
    if (chunk >= 0) {
      int b = chunk * 64 + 63;
      for (; b > chunk * 64; --b) {
        const unsigned h = hist[b];
        if (acc + h >= (unsigned)TOPK_N) break;
        acc += h;
      }
      cut = ((unsigned)b) << 16;
    }
    *cutp = cut;
  }
}

// ---------------------------------------------------------------------------
// Pass 2: collect (valbits << 32) | ~flatidx keys for every value above cut.
// flatidx = ((s*2048 + r)*25 + i*5 + j) matches the reference flattening.
// ---------------------------------------------------------------------------
__global__ __launch_bounds__(256) void pair_collect_kernel(
    const float* __restrict__ aff,
    const int* __restrict__ sidx, const float* __restrict__ sdist,
    const int* __restrict__ ridx, const float* __restrict__ rdist,
    const unsigned* __restrict__ cutp, unsigned* __restrict__ counter,
    unsigned long long* __restrict__ cand) {
  const int gid = blockIdx.x * 256 + threadIdx.x;
  const int s = gid >> 11;
  const int r = gid & (N_PTS - 1);
  const float base = aff[gid];
  const unsigned cut = *cutp;

  float sd[KNB]; int so[KNB]; float rd[KNB]; int rj[KNB];
#pragma unroll
  for (int i = 0; i < KNB; ++i) {
    sd[i] = sdist[s * KNB + i];
    so[i] = sidx[s * KNB + i] << 11;
    rd[i] = rdist[r * KNB + i];
    rj[i] = ridx[r * KNB + i];
  }
  if (base <= 0.0f) return;
#pragma unroll
  for (int i = 0; i < KNB; ++i) {
#pragma unroll
    for (int j = 0; j < KNB; ++j) {
      const float del = sd[i] - rd[j];
      const float tmp = 1.0f - del * del * INV_S2;
      if (tmp > 0.0f) {
        const float v = base * tmp * aff[so[i] + rj[j]];
        if (v > 0.0f) {
          const unsigned vb = __float_as_uint(v);
          if (vb >= cut) {
            const unsigned pos = atomicAdd(counter, 1u);
            if (pos < CAND_CAP) {
              const unsigned flat =
                  (unsigned)gid * 25u + (unsigned)(i * KNB + j);
              cand[pos] =
                  ((unsigned long long)vb << 32) | (unsigned)(~flat);
            }
          }
        }
      }
    }
  }
}

// ---------------------------------------------------------------------------
// Final: 512 iterations of block-wide argmax over the candidate keys.
// Key order = (value desc, index asc) — matches jax.lax.top_k tie-breaking.
// Decodes flat index into the four output index arrays (concatenated).
// ---------------------------------------------------------------------------
__global__ __launch_bounds__(1024) void topk_select_kernel(
    unsigned long long* __restrict__ cand, const unsigned* __restrict__ counter,
    const int* __restrict__ sidx, const int* __restrict__ ridx,
    int* __restrict__ out) {
  __shared__ unsigned long long skey[1024];
  __shared__ int spos[1024];
  const int t = threadIdx.x;
  int n = (int)*counter;
  if (n > CAND_CAP) n = CAND_CAP;

  for (int it = 0; it < TOPK_N; ++it) {
    unsigned long long best = 0ull;
    int bpos = -1;
    for (int p = t; p < n; p += 1024) {
      const unsigned long long k = cand[p];
      if (k > best) { best = k; bpos = p; }
    }
    skey[t] = best;
    spos[t] = bpos;
    __syncthreads();
    for (int off = 512; off > 0; off >>= 1) {
      if (t < off && skey[t + off] > skey[t]) {
        skey[t] = skey[t + off];
        spos[t] = spos[t + off];
      }
      __syncthreads();
    }
    if (t == 0) {
      const unsigned long long k = skey[0];
      const unsigned flat =
          (k != 0ull) ? ~(unsigned)(k & 0xFFFFFFFFull) : 0u;
      const unsigned s = flat / (unsigned)FLATK;
      const unsigned rem = flat % (unsigned)FLATK;
      const unsigned r = rem / (unsigned)(KNB * KNB);
      const unsigned ij = rem % (unsigned)(KNB * KNB);
      out[it] = (int)s;                               // first_node_src
      out[TOPK_N + it] = (int)r;                      // first_node_ref
      out[2 * TOPK_N + it] = sidx[s * KNB + ij / KNB];// second_node_src
      out[3 * TOPK_N + it] = ridx[r * KNB + ij % KNB];// second_node_ref
      if (spos[0] >= 0) cand[spos[0]] = 0ull;         // mark taken
    }
    __syncthreads();
  }
}

// ---------------------------------------------------------------------------
extern "C" void kernel_launch(void* const* d_in, const int* in_sizes, int n_in,
                              void* d_out, int out_size, void* d_ws,
                              size_t ws_size, hipStream_t stream) {
  (void)in_sizes; (void)n_in; (void)out_size; (void)ws_size;
  const float* src = (const float*)d_in[0];   // [2048,3]
  const float* ref = (const float*)d_in[1];   // [2048,3]
  const float* aff = (const float*)d_in[2];   // [2048,2048]
  int* out = (int*)d_out;                     // 4 x 512 int32, concatenated

  // workspace layout (all 8B-aligned): hist | counter | cut | pad |
  //   s_idx | s_dist | r_idx | r_dist | cand
  unsigned* hist    = (unsigned*)d_ws;
  unsigned* counter = hist + 65536;
  unsigned* cutp    = hist + 65537;
  int*   s_idx  = (int*)(hist + 65544);
  float* s_dist = (float*)(s_idx + N_PTS * KNB);
  int*   r_idx  = (int*)(s_dist + N_PTS * KNB);
  float* r_dist = (float*)(r_idx + N_PTS * KNB);
  unsigned long long* cand = (unsigned long long*)(r_dist + N_PTS * KNB);

  init_ws_kernel<<<(65538 + 255) / 256, 256, 0, stream>>>(hist, 65538);
  knn_wmma_kernel<<<N_PTS / 16, 32, 0, stream>>>(src, s_idx, s_dist);
  knn_wmma_kernel<<<N_PTS / 16, 32, 0, stream>>>(ref, r_idx, r_dist);
  pair_hist_kernel<<<(N_PTS * N_PTS) / 256, 256, 0, stream>>>(
      aff, s_idx, s_dist, r_idx, r_dist, hist);
  hist_scan_kernel<<<1, 1024, 0, stream>>>(hist, cutp);
  pair_collect_kernel<<<(N_PTS * N_PTS) / 256, 256, 0, stream>>>(
      aff, s_idx, s_dist, r_idx, r_dist, cutp, counter, cand);
  topk_select_kernel<<<1, 1024, 0, stream>>>(cand, counter, s_idx, r_idx, out);
}